// RNN_Cell_86912958202398
// MI455X (gfx1250) — compile-verified
//
#include <hip/hip_runtime.h>
#include <hip/hip_bf16.h>
#include <math.h>
#include <stdint.h>

// Problem constants (match reference)
#define T_STEPS 512
#define BATCH   64
#define DIM     1024   // D == H == 1024

typedef float v2f __attribute__((ext_vector_type(2)));
typedef float v4f __attribute__((ext_vector_type(4)));
typedef float v8f __attribute__((ext_vector_type(8)));
typedef unsigned int u32x4 __attribute__((ext_vector_type(4)));
typedef int i32x4 __attribute__((ext_vector_type(4)));
typedef int i32x8 __attribute__((ext_vector_type(8)));

// ---------------------------------------------------------------------------
// V_WMMA_F32_16X16X4_F32 operand layouts (wave32):
//   A 16x4 : lane = m + 16*(k>>1), vgpr = k&1 -> per-lane float2 {K=2h, K=2h+1}
//   B 4x16 : lane = n + 16*(k>>1), vgpr = k&1
//   C/D    : vgpr r -> row (r + 8*(lane>>4)), col = lane&15
// ---------------------------------------------------------------------------
__device__ __forceinline__ v8f wmma4(v2f a, v2f b, v8f c) {
    return __builtin_amdgcn_wmma_f32_16x16x4_f32(false, a, false, b,
                                                 (short)0, c, false, false);
}

// ---------------------------------------------------------------------------
// Tensor Data Mover: DMA a 2-D fp32 tile (tile_w x tile_h elements, row
// stride = stride_elems in memory) into LDS at byte offset lds_off, adding
// pad_amt (code: DWORDs-1) after every pad_int (code: 2<<code DWORDs) to
// shape the LDS row stride for bank-conflict-free fragment reads.
// Descriptor per CDNA5 ISA (count=1, type=2, 4B data).
// This toolchain's builtin is the 6-arg clang-23 form:
//   (u32x4 g0, i32x8 g1, i32x4 g2, i32x4 g3, i32x8 extra, i32 cpol)
// groups 2/3/extra are zero for a 2-D tile (tile_dim2..4 = 0).
// ---------------------------------------------------------------------------
__device__ __forceinline__ void tdm_load_2d(const float* gsrc, unsigned lds_off,
                                            unsigned stride_elems,
                                            unsigned tile_w, unsigned tile_h,
                                            unsigned pad_int_code,
                                            unsigned pad_amt_code)
{
    unsigned long long ga = (unsigned long long)(uintptr_t)gsrc;
    u32x4 g0;
    g0[0] = 1u;                                   // count=1, user descriptor
    g0[1] = lds_off;                              // lds_addr [63:32]
    g0[2] = (unsigned)ga;                         // global_addr lo
    g0[3] = ((unsigned)(ga >> 32) & 0x01FFFFFFu)  // global_addr [56:32]
            | (2u << 30);                         // type = 2 ("image")
    i32x8 g1;
    g1[0] = (int)((2u << 16)                      // data_size = 4 bytes
                | (1u << 20)                      // pad_enable
                | (pad_int_code << 22)
                | (pad_amt_code << 25));
    g1[1] = (int)((tile_w & 0xFFFFu) << 16);      // tensor_dim0 lo16 (abar=0)
    g1[2] = (int)(((tile_w >> 16) & 0xFFFFu) | ((tile_h & 0xFFFFu) << 16));
    g1[3] = (int)(((tile_h >> 16) & 0xFFFFu) | (tile_w << 16));   // tile_dim0
    g1[4] = (int)(tile_h & 0xFFFFu);              // tile_dim1 (tile_dim2 = 0)
    g1[5] = (int)stride_elems;                    // tensor_dim0_stride lo32
    g1[6] = 0;
    g1[7] = 0;
    i32x4 z4 = {0, 0, 0, 0};
    i32x8 z8 = {0, 0, 0, 0, 0, 0, 0, 0};
    __builtin_amdgcn_tensor_load_to_lds(g0, g1, z4, z4, z8, 0);
}

// ---------------------------------------------------------------------------
// Phase 1: xproj = inputs[32768,1024] @ W_xh[1024,1024] + b_h   (-> d_out)
// Block = 256 thr (8 waves) computes a 128M x 64N tile; K chunked by 64.
// A chunk (128x64) and B chunk (64x64) staged in LDS by TDM, double-buffered.
// LDS strides: A rows 68 floats (pad 4 DW / 64 DW), B rows 72 (pad 8 DW).
// Each wave: 16 M-rows x 64 N-cols = 4 accumulators, A fragment reused 4x.
// ---------------------------------------------------------------------------
#define AS_STRIDE 68
#define BS_STRIDE 72
#define AS_BYTES  (128 * AS_STRIDE * 4)   // 34816
#define BS_BYTES  (64  * BS_STRIDE * 4)   // 18432
#define P1_SMEM   (2 * (AS_BYTES + BS_BYTES))   // 106496

__global__ void __launch_bounds__(256)
rnn_xproj_kernel(const float* __restrict__ X,
                 const float* __restrict__ W,
                 const float* __restrict__ bias,
                 float* __restrict__ O)
{
    extern __shared__ char smem[];
    // buffer layout: As0 | As1 | Bs0 | Bs1
    const unsigned asOff[2] = {0u, (unsigned)AS_BYTES};
    const unsigned bsOff[2] = {2u * AS_BYTES, 2u * AS_BYTES + BS_BYTES};
    const unsigned ldsBase = (unsigned)(uintptr_t)&smem[0];   // LDS byte addr

    const int lane = threadIdx.x & 31;
    const int wave = threadIdx.x >> 5;
    const int idx  = lane & 15;
    const int half = lane >> 4;

    const int mBase = (blockIdx.x >> 4) * 128;   // 256 M-blocks
    const int n0blk = (blockIdx.x & 15) * 64;    // 16 N-blocks

    const int nChunks = DIM / 64;                // 16

    // prologue: chunk 0 into buffer 0 (wave 0 drives the TDM)
    if (threadIdx.x < 32) {
        tdm_load_2d(X + (size_t)mBase * DIM, ldsBase + asOff[0], DIM, 64, 128, 5, 3);
        tdm_load_2d(W + n0blk,               ldsBase + bsOff[0], DIM, 64, 64, 5, 7);
    }

    v8f acc[4] = {v8f{}, v8f{}, v8f{}, v8f{}};

    for (int c = 0; c < nChunks; ++c) {
        const int buf = c & 1;
        if (threadIdx.x < 32) {
            if (c + 1 < nChunks) {   // prefetch next chunk into other buffer
                const int kc = (c + 1) * 64;
                tdm_load_2d(X + (size_t)mBase * DIM + kc, ldsBase + asOff[buf ^ 1],
                            DIM, 64, 128, 5, 3);
                tdm_load_2d(W + (size_t)kc * DIM + n0blk, ldsBase + bsOff[buf ^ 1],
                            DIM, 64, 64, 5, 7);
                __builtin_amdgcn_s_wait_tensorcnt(2);  // chunk c complete
            } else {
                __builtin_amdgcn_s_wait_tensorcnt(0);
            }
        }
        __syncthreads();

        const float* As = (const float*)(smem + asOff[buf]);
        const float* Bs = (const float*)(smem + bsOff[buf]);
        const float* arow = As + (wave * 16 + idx) * AS_STRIDE;

#pragma unroll
        for (int kk = 0; kk < 64; kk += 4) {
            const int ka = kk + 2 * half;
            v2f a = *(const v2f*)(arow + ka);           // ds_load_b64
#pragma unroll
            for (int nt = 0; nt < 4; ++nt) {
                v2f b;
                b.x = Bs[(ka + 0) * BS_STRIDE + nt * 16 + idx];
                b.y = Bs[(ka + 1) * BS_STRIDE + nt * 16 + idx];
                acc[nt] = wmma4(a, b, acc[nt]);
            }
        }
        __syncthreads();   // chunk consumed; buffer may be overwritten
    }

    // epilogue: add bias, store 4 tiles
    const int grow = mBase + wave * 16 + 8 * half;
#pragma unroll
    for (int nt = 0; nt < 4; ++nt) {
        const int col = n0blk + nt * 16 + idx;
        const float bv = bias[col];
#pragma unroll
        for (int r = 0; r < 8; ++r) {
            O[(size_t)(grow + r) * DIM + col] = acc[nt][r] + bv;
        }
    }
}

// ---------------------------------------------------------------------------
// Phase 2: persistent scan kernel.  32 blocks x 8 waves = 256 waves; wave w
// owns one 16x16 tile of h (4 M-tiles x 64 N-tiles).  Steps t = 0..511 are
// separated by an L2 atomic barrier (monotonic counter in d_ws).  Two
// interleaved accumulators halve the dependent-WMMA issue interval.
// ---------------------------------------------------------------------------
__global__ void __launch_bounds__(256)
rnn_scan_kernel(const float* __restrict__ state,   // [64,1024] h_{-1}
                const float* __restrict__ W,       // [1024,1024]
                float* __restrict__ out,           // [T,64,1024]; holds xproj
                unsigned* __restrict__ ctr)        // zeroed barrier counter
{
    const int lane = threadIdx.x & 31;
    const int idx  = lane & 15;
    const int half = lane >> 4;

    const int gw = blockIdx.x * 8 + (threadIdx.x >> 5);  // 0..255
    const int m0 = (gw & 3) * 16;                        // M fastest
    const int n0 = (gw >> 2) * 16;

    const float* bcol = W + n0 + idx;
    const unsigned nBlocks = gridDim.x;

    for (int t = 0; t < T_STEPS; ++t) {
        const float* hp = (t == 0) ? state
                                   : out + (size_t)(t - 1) * BATCH * DIM;
        const float* arow = hp + (size_t)(m0 + idx) * DIM;

        v8f acc0 = {}, acc1 = {};
#pragma unroll 4
        for (int kk = 0; kk < DIM; kk += 8) {
            const int ka = kk + 2 * half;
            v2f a0 = *(const v2f*)(arow + ka);
            v2f b0;
            b0.x = bcol[(size_t)(ka + 0) * DIM];
            b0.y = bcol[(size_t)(ka + 1) * DIM];
            acc0 = wmma4(a0, b0, acc0);

            const int kb = ka + 4;
            v2f a1 = *(const v2f*)(arow + kb);
            v2f b1;
            b1.x = bcol[(size_t)(kb + 0) * DIM];
            b1.y = bcol[(size_t)(kb + 1) * DIM];
            acc1 = wmma4(a1, b1, acc1);
        }

        float* ot = out + (size_t)t * BATCH * DIM;
        const int col = n0 + idx;
#pragma unroll
        for (int r = 0; r < 8; ++r) {
            const size_t o = (size_t)(m0 + r + 8 * half) * DIM + col;
            ot[o] = tanhf(ot[o] + acc0[r] + acc1[r]);
        }

        // ---- device-wide step barrier (release -> count -> acquire) -------
        __threadfence();
        __syncthreads();
        if (threadIdx.x == 0) {
            __hip_atomic_fetch_add(ctr, 1u, __ATOMIC_RELAXED,
                                   __HIP_MEMORY_SCOPE_AGENT);
            const unsigned target = (unsigned)(t + 1) * nBlocks;
            while (__hip_atomic_load(ctr, __ATOMIC_RELAXED,
                                     __HIP_MEMORY_SCOPE_AGENT) < target) {
                __builtin_amdgcn_s_sleep(2);
            }
        }
        __syncthreads();
        __threadfence();   // acquire: next step sees other blocks' h writes
    }
}

extern "C" void kernel_launch(void* const* d_in, const int* in_sizes, int n_in,
                              void* d_out, int out_size, void* d_ws, size_t ws_size,
                              hipStream_t stream)
{
    (void)in_sizes; (void)n_in; (void)out_size; (void)ws_size;

    const float* inputs = (const float*)d_in[0];  // [T,B,D]
    const float* state  = (const float*)d_in[1];  // [B,H]
    const float* W_xh   = (const float*)d_in[2];  // [D,H]
    const float* W_hh   = (const float*)d_in[3];  // [H,H]
    const float* b_h    = (const float*)d_in[4];  // [H]

    float* out         = (float*)d_out;                        // [T,B,H]
    float* final_state = out + (size_t)T_STEPS * BATCH * DIM;  // [B,H]

    // Phase 1: TDM-fed LDS-tiled GEMM, xproj written in place into d_out.
    {
        const int blocks = (T_STEPS * BATCH / 128) * (DIM / 64);  // 4096
        rnn_xproj_kernel<<<blocks, 256, P1_SMEM, stream>>>(inputs, W_xh, b_h, out);
    }

    // Phase 2: persistent scan with device-wide atomic step barrier.
    {
        (void)hipMemsetAsync(d_ws, 0, 128, stream);   // barrier counter := 0
        rnn_scan_kernel<<<32, 256, 0, stream>>>(state, W_hh, out,
                                                (unsigned*)d_ws);
    }

    // Final state = last timestep.
    (void)hipMemcpyAsync(final_state, out + (size_t)(T_STEPS - 1) * BATCH * DIM,
                         (size_t)BATCH * DIM * sizeof(float),
                         hipMemcpyDeviceToDevice, stream);
}